// ServiceFailureGAT_32839319945416
// MI455X (gfx1250) — compile-verified
//
#include <hip/hip_runtime.h>
#include <hip/hip_bf16.h>
#include <math.h>

typedef float v2f __attribute__((ext_vector_type(2)));
typedef float v8f __attribute__((ext_vector_type(8)));

#define NEG_SLOPE 0.2f

__device__ __forceinline__ float lrelu(float x) { return x > 0.f ? x : NEG_SLOPE * x; }

// Monotone float <-> int key mapping for atomicMax-based segment max
__device__ __forceinline__ int f2key(float f) {
    int i = __float_as_int(f);
    return (i >= 0) ? i : (i ^ 0x7FFFFFFF);
}
__device__ __forceinline__ float key2f(int i) {
    return __int_as_float((i >= 0) ? i : (i ^ 0x7FFFFFFF));
}

__device__ __forceinline__ void edge_sd(const int* __restrict__ ei, int E, int eid, int& s, int& d) {
    if (eid < E) { s = ei[eid]; d = ei[E + eid]; }
    else         { s = d = eid - E; }          // self loops appended
}

// ---------------- init ----------------
__global__ void fill_i32(int* p, int v, size_t n) {
    size_t i = (size_t)blockIdx.x * blockDim.x + threadIdx.x;
    if (i < n) p[i] = v;
}

// ---------------- layer-1 GEMM: H1[N,128] = X[N,128] @ W1[128,128] ----------------
// One wave computes one 16x16 output tile via V_WMMA_F32_16X16X4_F32, K-loop of 32 steps.
__global__ __launch_bounds__(256) void gemm1_wmma(const float* __restrict__ X,
                                                  const float* __restrict__ W,
                                                  float* __restrict__ H, int n) {
    int lane = threadIdx.x & 31;
    int wave = threadIdx.x >> 5;          // 8 waves -> 8 col tiles (128 cols)
    int tm   = blockIdx.x;                // row tile
    int kh   = lane >> 4;                 // 0 -> K {0,1}, 1 -> K {2,3}
    int l16  = lane & 15;

    int row  = tm * 16 + l16;
    int arow = row < n ? row : (n - 1);   // clamp for safety (n%16==0 in practice)
    int col  = wave * 16 + l16;

    const float* xr = X + (size_t)arow * 128;
    v8f c = {};
    #pragma unroll 4
    for (int k0 = 0; k0 < 128; k0 += 4) {
        int ka = k0 + kh * 2;
        v2f a; a.x = xr[ka];                       a.y = xr[ka + 1];
        v2f b; b.x = W[(size_t)ka * 128 + col];    b.y = W[(size_t)(ka + 1) * 128 + col];
        c = __builtin_amdgcn_wmma_f32_16x16x4_f32(false, a, false, b,
                                                  (short)0, c, false, false);
    }
    // C/D layout: VGPR r -> M = r + 8*kh ; N = l16
    #pragma unroll
    for (int r = 0; r < 8; ++r) {
        int orow = tm * 16 + kh * 8 + r;
        if (orow < n) H[(size_t)orow * 128 + col] = c[r];
    }
}

// ---------------- layer-1 attention coefficients per (node, head) ----------------
__global__ void alpha1_k(const float* __restrict__ H, const float* __restrict__ a_src,
                         const float* __restrict__ a_dst, float* __restrict__ as,
                         float* __restrict__ ad, int n) {
    int i = blockIdx.x * blockDim.x + threadIdx.x;  // over n*2
    if (i >= n * 2) return;
    int node = i >> 1, h = i & 1;
    const float* hp = H + (size_t)node * 128 + h * 64;
    const float* sp = a_src + h * 64;
    const float* dp = a_dst + h * 64;
    float ss = 0.f, dd = 0.f;
    #pragma unroll 8
    for (int cidx = 0; cidx < 64; ++cidx) { float v = hp[cidx]; ss += v * sp[cidx]; dd += v * dp[cidx]; }
    as[i] = ss; ad[i] = dd;
}

// ---------------- layer-1 edge passes ----------------
__global__ void edge1_max(const int* __restrict__ ei, int E, int Etot,
                          const float* __restrict__ as, const float* __restrict__ ad,
                          int* __restrict__ m) {
    int i = blockIdx.x * blockDim.x + threadIdx.x;  // over Etot*2
    if (i >= Etot * 2) return;
    int eid = i >> 1, h = i & 1, s, d;
    edge_sd(ei, E, eid, s, d);
    float e = lrelu(as[s * 2 + h] + ad[d * 2 + h]);
    atomicMax(&m[d * 2 + h], f2key(e));
}

__global__ void edge1_expsum(const int* __restrict__ ei, int E, int Etot,
                             const float* __restrict__ as, const float* __restrict__ ad,
                             const int* __restrict__ m, float* __restrict__ ex,
                             float* __restrict__ den) {
    int i = blockIdx.x * blockDim.x + threadIdx.x;  // over Etot*2
    if (i >= Etot * 2) return;
    int eid = i >> 1, h = i & 1, s, d;
    edge_sd(ei, E, eid, s, d);
    float e  = lrelu(as[s * 2 + h] + ad[d * 2 + h]);
    float mx = key2f(m[d * 2 + h]);
    float v  = expf(e - mx);
    ex[i] = v;
    atomicAdd(&den[d * 2 + h], v);
}

// 128 threads per edge (one feature each), 2 edges per 256-thread block
__global__ __launch_bounds__(256) void edge1_scatter(const int* __restrict__ ei, int E, int Etot,
                                                     const float* __restrict__ H,
                                                     const float* __restrict__ ex,
                                                     const float* __restrict__ den,
                                                     float* __restrict__ acc) {
    int eid = blockIdx.x * 2 + (threadIdx.x >> 7);
    if (eid >= Etot) return;
    int f = threadIdx.x & 127;
    int h = f >> 6;
    int s, d;
    edge_sd(ei, E, eid, s, d);
    float alpha = ex[(size_t)eid * 2 + h] / (den[d * 2 + h] + 1e-16f);
    atomicAdd(&acc[(size_t)d * 128 + f], H[(size_t)s * 128 + f] * alpha);
}

// ---------------- layer-1 finalize + layer-2 node projection (one wave per node) ----------------
__global__ __launch_bounds__(256) void node1_fin(const float* __restrict__ acc,
                                                 const float* __restrict__ b1,
                                                 const float* __restrict__ W2,
                                                 const float* __restrict__ a_src2,
                                                 const float* __restrict__ a_dst2,
                                                 float* __restrict__ h2,
                                                 float* __restrict__ as2,
                                                 float* __restrict__ ad2, int n) {
    int wave = threadIdx.x >> 5;
    int lane = threadIdx.x & 31;
    int node = blockIdx.x * 8 + wave;
    if (node >= n) return;
    const float* ap = acc + (size_t)node * 128;
    float sum = 0.f;
    #pragma unroll
    for (int j = 0; j < 4; ++j) {
        int f = lane + j * 32;
        float v = ap[f] + b1[f];
        v = v > 0.f ? v : 0.f;      // ReLU between layers
        sum += v * W2[f];
    }
    #pragma unroll
    for (int off = 16; off > 0; off >>= 1) sum += __shfl_xor(sum, off, 32);
    if (lane == 0) {
        h2[node]  = sum;
        as2[node] = sum * a_src2[0];
        ad2[node] = sum * a_dst2[0];
    }
}

// ---------------- layer-2 edge passes (1 head, 1 channel) ----------------
__global__ void edge2_max(const int* __restrict__ ei, int E, int Etot,
                          const float* __restrict__ as, const float* __restrict__ ad,
                          int* __restrict__ m) {
    int eid = blockIdx.x * blockDim.x + threadIdx.x;
    if (eid >= Etot) return;
    int s, d;
    edge_sd(ei, E, eid, s, d);
    atomicMax(&m[d], f2key(lrelu(as[s] + ad[d])));
}

__global__ void edge2_expsum(const int* __restrict__ ei, int E, int Etot,
                             const float* __restrict__ as, const float* __restrict__ ad,
                             const int* __restrict__ m, float* __restrict__ ex,
                             float* __restrict__ den) {
    int eid = blockIdx.x * blockDim.x + threadIdx.x;
    if (eid >= Etot) return;
    int s, d;
    edge_sd(ei, E, eid, s, d);
    float e = lrelu(as[s] + ad[d]);
    float v = expf(e - key2f(m[d]));
    ex[eid] = v;
    atomicAdd(&den[d], v);
}

__global__ void edge2_scatter(const int* __restrict__ ei, int E, int Etot,
                              const float* __restrict__ h2,
                              const float* __restrict__ ex,
                              const float* __restrict__ den,
                              float* __restrict__ out2) {
    int eid = blockIdx.x * blockDim.x + threadIdx.x;
    if (eid >= Etot) return;
    int s, d;
    edge_sd(ei, E, eid, s, d);
    float alpha = ex[eid] / (den[d] + 1e-16f);
    atomicAdd(&out2[d], h2[s] * alpha);
}

__global__ void final_sigmoid(const float* __restrict__ out2, const float* __restrict__ b2,
                              float* __restrict__ out, int n) {
    int i = blockIdx.x * blockDim.x + threadIdx.x;
    if (i >= n) return;
    float v = out2[i] + b2[0];
    out[i] = 1.f / (1.f + expf(-v));
}

extern "C" void kernel_launch(void* const* d_in, const int* in_sizes, int n_in,
                              void* d_out, int out_size, void* d_ws, size_t ws_size,
                              hipStream_t stream) {
    const float* x      = (const float*)d_in[0];
    const int*   ei     = (const int*)d_in[1];
    const float* W1     = (const float*)d_in[2];
    const float* a_src1 = (const float*)d_in[3];
    const float* a_dst1 = (const float*)d_in[4];
    const float* b1     = (const float*)d_in[5];
    const float* W2     = (const float*)d_in[6];
    const float* a_src2 = (const float*)d_in[7];
    const float* a_dst2 = (const float*)d_in[8];
    const float* b2     = (const float*)d_in[9];
    float* out = (float*)d_out;

    const int n    = in_sizes[0] / 128;
    const int E    = in_sizes[1] / 2;
    const int Etot = E + n;                  // + self loops

    // workspace layout
    float* H1   = (float*)d_ws;                       // n*128
    float* as1  = H1   + (size_t)n * 128;             // n*2
    float* ad1  = as1  + (size_t)n * 2;               // n*2
    int*   m1   = (int*)(ad1 + (size_t)n * 2);        // n*2
    float* den1 = (float*)(m1 + (size_t)n * 2);       // n*2
    float* ex1  = den1 + (size_t)n * 2;               // Etot*2
    float* acc1 = ex1  + (size_t)Etot * 2;            // n*128
    float* h2   = acc1 + (size_t)n * 128;             // n
    float* as2  = h2   + n;                           // n
    float* ad2  = as2  + n;                           // n
    int*   m2   = (int*)(ad2 + n);                    // n
    float* den2 = (float*)(m2 + n);                   // n
    float* ex2  = den2 + n;                           // Etot
    float* out2 = ex2  + Etot;                        // n

    const int T = 256;
    auto blocks = [](size_t cnt, int t) { return (unsigned)((cnt + t - 1) / t); };

    // init: max-keys to INT_MIN, accumulators to zero
    fill_i32<<<blocks((size_t)n * 2, T), T, 0, stream>>>(m1, INT_MIN, (size_t)n * 2);
    fill_i32<<<blocks((size_t)n, T), T, 0, stream>>>(m2, INT_MIN, (size_t)n);
    fill_i32<<<blocks((size_t)n * 2, T), T, 0, stream>>>((int*)den1, 0, (size_t)n * 2);
    fill_i32<<<blocks((size_t)n * 128, T), T, 0, stream>>>((int*)acc1, 0, (size_t)n * 128);
    fill_i32<<<blocks((size_t)n, T), T, 0, stream>>>((int*)den2, 0, (size_t)n);
    fill_i32<<<blocks((size_t)n, T), T, 0, stream>>>((int*)out2, 0, (size_t)n);

    // layer 1
    gemm1_wmma<<<(n + 15) / 16, 256, 0, stream>>>(x, W1, H1, n);
    alpha1_k<<<blocks((size_t)n * 2, T), T, 0, stream>>>(H1, a_src1, a_dst1, as1, ad1, n);
    edge1_max<<<blocks((size_t)Etot * 2, T), T, 0, stream>>>(ei, E, Etot, as1, ad1, m1);
    edge1_expsum<<<blocks((size_t)Etot * 2, T), T, 0, stream>>>(ei, E, Etot, as1, ad1, m1, ex1, den1);
    edge1_scatter<<<(Etot + 1) / 2, 256, 0, stream>>>(ei, E, Etot, H1, ex1, den1, acc1);

    // finalize layer 1 + layer-2 projection
    node1_fin<<<(n + 7) / 8, 256, 0, stream>>>(acc1, b1, W2, a_src2, a_dst2, h2, as2, ad2, n);

    // layer 2
    edge2_max<<<blocks((size_t)Etot, T), T, 0, stream>>>(ei, E, Etot, as2, ad2, m2);
    edge2_expsum<<<blocks((size_t)Etot, T), T, 0, stream>>>(ei, E, Etot, as2, ad2, m2, ex2, den2);
    edge2_scatter<<<blocks((size_t)Etot, T), T, 0, stream>>>(ei, E, Etot, h2, ex2, den2, out2);

    final_sigmoid<<<blocks((size_t)n, T), T, 0, stream>>>(out2, b2, out, n);
}